// GNN_drug_3066606649962
// MI455X (gfx1250) — compile-verified
//
#include <hip/hip_runtime.h>

typedef __attribute__((ext_vector_type(2))) float v2f;
typedef __attribute__((ext_vector_type(8))) float v8f;

#define N_NODES 300000
#define N_EDGES 600000
#define N_GRAPHS 6000
#define DIM 128
#define IN_DIM 77
#define BN_EPS 1e-5f
#define LDS_STRIDE 132   // 128 + 4 pad -> conflict-free ds_load_b64
#define TILES 3          // row-tiles of 16 per block
#define ROWS (16 * TILES)  // 48; 300000 % 48 == 0

// ---------------------------------------------------------------------------
// Transpose all 6 weight matrices into wt[m][n][k] = W_m[k][n], K zero-padded
// ---------------------------------------------------------------------------
__global__ void wt_transpose_kernel(const float* __restrict__ w0, const float* __restrict__ w1,
                                    const float* __restrict__ w2, const float* __restrict__ w3,
                                    const float* __restrict__ w4, const float* __restrict__ w5,
                                    float* __restrict__ wt) {
    int tid = blockIdx.x * blockDim.x + threadIdx.x;
    if (tid >= 6 * DIM * DIM) return;
    int m   = tid / (DIM * DIM);
    int rem = tid - m * DIM * DIM;
    int n   = rem / DIM;      // output column (row of wt)
    int k   = rem - n * DIM;  // reduction index
    const float* W; int fan;
    switch (m) {
        case 0:  W = w0; fan = IN_DIM; break;
        case 1:  W = w1; fan = DIM;    break;
        case 2:  W = w2; fan = DIM;    break;
        case 3:  W = w3; fan = DIM;    break;
        case 4:  W = w4; fan = DIM;    break;
        default: W = w5; fan = DIM;    break;
    }
    wt[tid] = (k < fan) ? W[k * DIM + n] : 0.0f;
}

// ---------------------------------------------------------------------------
// z[i][d] = h_in[i][d] for d < in_dim else 0  (z has fixed stride 128)
// ---------------------------------------------------------------------------
__global__ void init_z_kernel(float* __restrict__ z, const float* __restrict__ h,
                              int in_dim, int stride_in) {
    size_t tid = (size_t)blockIdx.x * blockDim.x + threadIdx.x;
    if (tid >= (size_t)N_NODES * DIM) return;
    int d = (int)(tid & (DIM - 1));
    size_t i = tid >> 7;
    z[tid] = (d < in_dim) ? h[i * (size_t)stride_in + d] : 0.0f;
}

// ---------------------------------------------------------------------------
// z[dst] += h[src] for every edge (fp32 atomics, 4 dims per thread)
// ---------------------------------------------------------------------------
__global__ void scatter_kernel(float* __restrict__ z, const float* __restrict__ h,
                               const int* __restrict__ ei, int in_dim, int stride_in,
                               int nchunk) {
    int tid = blockIdx.x * blockDim.x + threadIdx.x;
    if (tid >= N_EDGES * nchunk) return;
    int e = tid / nchunk;
    int c = tid - e * nchunk;
    int src = ei[e];
    int dst = ei[N_EDGES + e];
    int d0 = c * 4;
    const float* hp = h + (size_t)src * stride_in;
    float*       zp = z + (size_t)dst * DIM;
#pragma unroll
    for (int j = 0; j < 4; ++j) {
        int d = d0 + j;
        if (d < in_dim) atomicAdd(zp + d, hp[d]);
    }
}

// ---------------------------------------------------------------------------
// out = relu(zin @ W + b) via V_WMMA_F32_16X16X4_F32.
// D = Wt_tile(16x4) x Zt_tile(4x16):  D[m][n] -> out[node + n][col c0 + m]
// Block = 256 thr = 8 waves; block covers 48 rows (3 row-tiles) x 128 cols.
// Each wave: one weight v2f load feeds 3 WMMAs (3 independent acc chains).
// Safe in-place (zin == out): input tile fully staged to LDS before stores.
// ---------------------------------------------------------------------------
__global__ __launch_bounds__(256) void gin_gemm_kernel(const float* zin,
                                                       const float* __restrict__ wt,
                                                       const float* __restrict__ bias,
                                                       float* out, int kmax) {
    __shared__ float sz[ROWS * LDS_STRIDE];
    const int r0  = blockIdx.x * ROWS;
    const int tid = threadIdx.x;

    // cooperative stage: 48 rows x 128 floats = 1536 float4, 6 per thread
#pragma unroll
    for (int j = 0; j < (ROWS * DIM / 4) / 256; ++j) {
        int idx4 = tid + j * 256;               // 0..1535
        int row  = idx4 >> 5;
        int c4   = (idx4 & 31) * 4;
        float4 v = *(const float4*)(zin + (size_t)(r0 + row) * DIM + c4);
        *(float4*)(&sz[row * LDS_STRIDE + c4]) = v;
    }
    __syncthreads();

    const int lane = tid & 31;
    const int wave = tid >> 5;
    const int c0   = wave * 16;        // this wave's output-column tile
    const int half = lane >> 4;        // 0: K pair {0,1}, 1: K pair {2,3}
    const int l    = lane & 15;

    const float* arow = wt + (size_t)(c0 + l) * DIM + 2 * half;   // Wt[c0+l][k..]
    const float* brow = sz + l * LDS_STRIDE + 2 * half;           // z[r0+l][k..]

    v8f acc[TILES];
    {
        const float4* bp = (const float4*)(bias + c0 + half * 8);
        float4 b0 = bp[0], b1 = bp[1];
        v8f binit = (v8f){b0.x, b0.y, b0.z, b0.w, b1.x, b1.y, b1.z, b1.w};
#pragma unroll
        for (int t = 0; t < TILES; ++t) acc[t] = binit;
    }

#pragma unroll 2
    for (int k0 = 0; k0 < kmax; k0 += 4) {
        v2f a = *(const v2f*)(arow + k0);       // shared across all 3 row-tiles
#pragma unroll
        for (int t = 0; t < TILES; ++t) {
            v2f b = *(const v2f*)(brow + t * 16 * LDS_STRIDE + k0);
            acc[t] = __builtin_amdgcn_wmma_f32_16x16x4_f32(false, a, false, b,
                                                           (short)0, acc[t], false, false);
        }
    }

#pragma unroll
    for (int t = 0; t < TILES; ++t) {
        float* op = out + (size_t)(r0 + 16 * t + l) * DIM + c0 + half * 8;
        float4 o0, o1;
        o0.x = fmaxf(acc[t][0], 0.0f);
        o0.y = fmaxf(acc[t][1], 0.0f);
        o0.z = fmaxf(acc[t][2], 0.0f);
        o0.w = fmaxf(acc[t][3], 0.0f);
        o1.x = fmaxf(acc[t][4], 0.0f);
        o1.y = fmaxf(acc[t][5], 0.0f);
        o1.z = fmaxf(acc[t][6], 0.0f);
        o1.w = fmaxf(acc[t][7], 0.0f);
        *(float4*)op       = o0;
        *(float4*)(op + 4) = o1;
    }
}

// ---------------------------------------------------------------------------
// BatchNorm: stats (sum, sumsq per column) then normalize in place
// ---------------------------------------------------------------------------
__global__ void zero_stats_kernel(float* stats) { stats[threadIdx.x] = 0.0f; }

__global__ void bn_stats_kernel(const float* __restrict__ h, float* __restrict__ stats) {
    __shared__ float shs[256], shq[256];
    int tid = threadIdx.x;
    int d = tid & 127;
    int r = tid >> 7;
    float s = 0.0f, q = 0.0f;
    for (int i = blockIdx.x * 2 + r; i < N_NODES; i += gridDim.x * 2) {
        float v = h[(size_t)i * DIM + d];
        s += v;
        q += v * v;
    }
    shs[tid] = s; shq[tid] = q;
    __syncthreads();
    if (tid < 128) {
        atomicAdd(&stats[d],       shs[tid] + shs[tid + 128]);
        atomicAdd(&stats[128 + d], shq[tid] + shq[tid + 128]);
    }
}

__global__ void bn_apply_kernel(float* __restrict__ h, const float* __restrict__ stats,
                                const float* __restrict__ gamma, const float* __restrict__ beta) {
    size_t tid = (size_t)blockIdx.x * blockDim.x + threadIdx.x;
    if (tid >= (size_t)N_NODES * DIM) return;
    int d = (int)(tid & (DIM - 1));
    const float inv_n = 1.0f / (float)N_NODES;
    float mu  = stats[d] * inv_n;
    float var = fmaxf(stats[128 + d] * inv_n - mu * mu, 0.0f);
    float inv = rsqrtf(var + BN_EPS);
    h[tid] = (h[tid] - mu) * inv * gamma[d] + beta[d];
}

// ---------------------------------------------------------------------------
// Per-graph segment-max over concat(h0,h1,h2); batch is sorted -> binary search
// ---------------------------------------------------------------------------
__global__ __launch_bounds__(384) void pool_kernel(const float* __restrict__ h0,
                                                   const float* __restrict__ h1,
                                                   const float* __restrict__ h2,
                                                   const int* __restrict__ batch,
                                                   float* __restrict__ out) {
    int g   = blockIdx.x;
    int col = threadIdx.x;            // 0..383
    int lo = 0, hi = N_NODES;
    while (lo < hi) { int mid = (lo + hi) >> 1; if (batch[mid] < g) lo = mid + 1; else hi = mid; }
    int start = lo;
    hi = N_NODES;
    while (lo < hi) { int mid = (lo + hi) >> 1; if (batch[mid] < g + 1) lo = mid + 1; else hi = mid; }
    int end = lo;
    const float* hb = (col < 128) ? h0 : ((col < 256) ? h1 : h2);
    int cc = col & 127;
    float m = -3.402823466e38f;
    for (int i = start; i < end; ++i)
        m = fmaxf(m, hb[(size_t)i * DIM + cc]);
    out[(size_t)g * 384 + col] = m;
}

// ---------------------------------------------------------------------------
extern "C" void kernel_launch(void* const* d_in, const int* in_sizes, int n_in,
                              void* d_out, int out_size, void* d_ws, size_t ws_size,
                              hipStream_t stream) {
    (void)in_sizes; (void)n_in; (void)out_size; (void)ws_size;
    const float* x     = (const float*)d_in[0];
    const float* w1_0  = (const float*)d_in[1];
    const float* b1_0  = (const float*)d_in[2];
    const float* w2_0  = (const float*)d_in[3];
    const float* b2_0  = (const float*)d_in[4];
    const float* w1_1  = (const float*)d_in[5];
    const float* b1_1  = (const float*)d_in[6];
    const float* w2_1  = (const float*)d_in[7];
    const float* b2_1  = (const float*)d_in[8];
    const float* w1_2  = (const float*)d_in[9];
    const float* b1_2  = (const float*)d_in[10];
    const float* w2_2  = (const float*)d_in[11];
    const float* b2_2  = (const float*)d_in[12];
    const float* gamma = (const float*)d_in[13];
    const float* beta  = (const float*)d_in[14];
    const int*   ei    = (const int*)d_in[15];
    const int*   batch = (const int*)d_in[16];
    float* out = (float*)d_out;

    const size_t NH = (size_t)N_NODES * DIM;
    float* ws    = (float*)d_ws;
    float* zA    = ws;
    float* h0    = zA + NH;
    float* h1    = h0 + NH;
    float* h2    = h1 + NH;
    float* wt    = h2 + NH;             // 6 * 128 * 128
    float* stats = wt + 6 * DIM * DIM;  // 256

    wt_transpose_kernel<<<(6 * DIM * DIM + 255) / 256, 256, 0, stream>>>(
        w1_0, w2_0, w1_1, w2_1, w1_2, w2_2, wt);

    const float* hin[3]  = {x, h0, h1};
    float*       hout[3] = {h0, h1, h2};
    const float* wt1[3]  = {wt,              wt + 2 * DIM * DIM, wt + 4 * DIM * DIM};
    const float* wt2[3]  = {wt + DIM * DIM,  wt + 3 * DIM * DIM, wt + 5 * DIM * DIM};
    const float* bb1[3]  = {b1_0, b1_1, b1_2};
    const float* bb2[3]  = {b2_0, b2_1, b2_2};
    const int indim[3]   = {IN_DIM, DIM, DIM};
    const int kceil[3]   = {80, DIM, DIM};   // K rounded up to multiple of 4 (zero-padded)

    const int nElem = N_NODES * DIM;
    for (int lyr = 0; lyr < 3; ++lyr) {
        int idim   = indim[lyr];
        int stride = idim;               // x rows stride 77, h rows stride 128
        init_z_kernel<<<(nElem + 255) / 256, 256, 0, stream>>>(zA, hin[lyr], idim, stride);
        int nchunk   = (idim + 3) / 4;
        int sThreads = N_EDGES * nchunk;
        scatter_kernel<<<(sThreads + 255) / 256, 256, 0, stream>>>(zA, hin[lyr], ei, idim,
                                                                   stride, nchunk);
        // MLP: zA <- relu(zA @ W1 + b1)  (in place), hout <- relu(zA @ W2 + b2)
        gin_gemm_kernel<<<N_NODES / ROWS, 256, 0, stream>>>(zA, wt1[lyr], bb1[lyr], zA, kceil[lyr]);
        gin_gemm_kernel<<<N_NODES / ROWS, 256, 0, stream>>>(zA, wt2[lyr], bb2[lyr], hout[lyr], DIM);
        zero_stats_kernel<<<1, 256, 0, stream>>>(stats);
        bn_stats_kernel<<<1024, 256, 0, stream>>>(hout[lyr], stats);
        bn_apply_kernel<<<(nElem + 255) / 256, 256, 0, stream>>>(hout[lyr], stats,
                                                                 gamma + lyr * DIM,
                                                                 beta + lyr * DIM);
    }
    pool_kernel<<<N_GRAPHS, 384, 0, stream>>>(h0, h1, h2, batch, out);
}